// LocalGrouper_86268713107575
// MI455X (gfx1250) — compile-verified
//
#include <hip/hip_runtime.h>
#include <hip/hip_bf16.h>

// Problem constants (from setup_inputs)
#define BB   8
#define NN   4096
#define CC   64
#define SS   1024
#define KK   32
#define CH   67          // C + 3
#define OUTC 131         // CH + C
#define CHUNK 256
#define NCHUNK (NN / CHUNK)
#define CAND (KK + CHUNK)  // 288
#define FLT_BIG 3.402823466e38f

typedef __attribute__((ext_vector_type(2))) float f32x2;
typedef __attribute__((ext_vector_type(8))) float f32x8;

// ---------------------------------------------------------------- utilities
__device__ __forceinline__ void waveArgMin(float& d, int& s) {
#pragma unroll
  for (int off = 16; off >= 1; off >>= 1) {
    float od = __shfl_xor(d, off, 32);
    int   os = __shfl_xor(s, off, 32);
    if (od < d || (od == d && os < s)) { d = od; s = os; }
  }
}

// ---------------------------------------------------------------- 0: zero stats
__global__ void init_stats_kernel(float* stats) {
  int t = threadIdx.x;
  if (t < 2 * BB) stats[t] = 0.0f;
}

// ---------------------------------------------------------------- 1: FPS
// one block per batch; sequential S-1 steps, parallel argmax over N in LDS
__global__ void fps_kernel(const float* __restrict__ xyz,
                           int* __restrict__ fps_idx,
                           float* __restrict__ new_xyz) {
  __shared__ float mind[NN];
  __shared__ float rv[256];
  __shared__ int   ri[256];
  const int b = blockIdx.x;
  const int tid = threadIdx.x;
  const float* xb = xyz + (size_t)b * NN * 3;

  const float c0x = xb[0], c0y = xb[1], c0z = xb[2];
  for (int n = tid; n < NN; n += 256) {
    float dx = xb[n * 3 + 0] - c0x;
    float dy = xb[n * 3 + 1] - c0y;
    float dz = xb[n * 3 + 2] - c0z;
    mind[n] = dx * dx + dy * dy + dz * dz;
  }
  if (tid == 0) fps_idx[(size_t)b * SS] = 0;
  __syncthreads();

  for (int s = 1; s < SS; ++s) {
    float best = -1.0f; int bi = 0;
    for (int n = tid; n < NN; n += 256) {
      float d = mind[n];
      if (d > best) { best = d; bi = n; }
    }
    rv[tid] = best; ri[tid] = bi;
    __syncthreads();
#pragma unroll
    for (int off = 128; off >= 1; off >>= 1) {
      if (tid < off) {
        if (rv[tid + off] > rv[tid] ||
            (rv[tid + off] == rv[tid] && ri[tid + off] < ri[tid])) {
          rv[tid] = rv[tid + off]; ri[tid] = ri[tid + off];
        }
      }
      __syncthreads();
    }
    int nxt = ri[0];
    if (tid == 0) fps_idx[(size_t)b * SS + s] = nxt;
    float cx = xb[nxt * 3 + 0], cy = xb[nxt * 3 + 1], cz = xb[nxt * 3 + 2];
    for (int n = tid; n < NN; n += 256) {
      float dx = xb[n * 3 + 0] - cx;
      float dy = xb[n * 3 + 1] - cy;
      float dz = xb[n * 3 + 2] - cz;
      float d = dx * dx + dy * dy + dz * dz;
      if (d < mind[n]) mind[n] = d;
    }
    __syncthreads();
  }

  // gather new_xyz -> d_out[0 : B*S*3)
  for (int s = tid; s < SS; s += 256) {
    int i = fps_idx[(size_t)b * SS + s];
    new_xyz[((size_t)b * SS + s) * 3 + 0] = xb[i * 3 + 0];
    new_xyz[((size_t)b * SS + s) * 3 + 1] = xb[i * 3 + 1];
    new_xyz[((size_t)b * SS + s) * 3 + 2] = xb[i * 3 + 2];
  }
}

// ---------------------------------------------------------------- 2: kNN via WMMA + async LDS staging
// grid = (S/16, B), 256 threads (8 wave32). Block owns 16 query rows.
// xyz chunks are double-buffered into LDS with GLOBAL_LOAD_ASYNC_TO_LDS_B32
// (ASYNCcnt), overlapping the copy of chunk c+1 with WMMA + top-K of chunk c.
// d2 tile = A(16x4) x B(4x16) via v_wmma_f32_16x16x4_f32 with
//   A row m = (-2x, -2y, -2z, |s|^2),  B col n = (x, y, z, 1)  -> |s|^2 - 2 s.n
// then add |n|^2 per column. Streaming top-K merge per 256-col chunk in LDS.
__global__ void knn_kernel(const float* __restrict__ xyz,
                           const float* __restrict__ new_xyz,
                           int* __restrict__ knn_idx) {
  __shared__ float cd[16][CAND];
  __shared__ int   ci[16][CAND];
  __shared__ float xbuf[2][CHUNK * 3];     // double-buffered chunk of xyz

  const int b   = blockIdx.y;
  const int s0  = blockIdx.x * 16;
  const int tid = threadIdx.x;
  const int lane = tid & 31;
  const int wave = tid >> 5;

  const float* xb = xyz + (size_t)b * NN * 3;
  const unsigned long long gbase = (unsigned long long)(uintptr_t)xb;

  // running top-K list (slots 0..K-1) starts at +inf
  for (int t = tid; t < 16 * KK; t += 256) {
    cd[t / KK][t % KK] = FLT_BIG;
    ci[t / KK][t % KK] = 0;
  }

  // A fragment: 16x4 f32, lanes 0-15 hold K=0/1, lanes 16-31 hold K=2/3
  const int m = lane & 15;
  const float* srow = new_xyz + ((size_t)b * SS + (s0 + m)) * 3;
  const float sx = srow[0], sy = srow[1], sz = srow[2];
  const float sn = sx * sx + sy * sy + sz * sz;
  f32x2 afrag;
  if (lane < 16) { afrag.x = -2.0f * sx; afrag.y = -2.0f * sy; }
  else           { afrag.x = -2.0f * sz; afrag.y = sn;         }

  // issue async copy of chunk 0 into xbuf[0]: 768 dwords, 3 per thread
#pragma unroll
  for (int q = 0; q < 3; ++q) {
    unsigned ldso = (unsigned)(uintptr_t)&xbuf[0][0] + (unsigned)(tid + q * 256) * 4u;
    unsigned voff = (unsigned)(tid + q * 256) * 4u;
    asm volatile("global_load_async_to_lds_b32 %0, %1, %2 offset:0"
                 :: "v"(ldso), "v"(voff), "s"(gbase) : "memory");
  }
  __syncthreads();

  for (int chunk = 0; chunk < NCHUNK; ++chunk) {
    const int nbase = chunk * CHUNK;
    const int buf = chunk & 1;

    // wait for this chunk's async copy, then make it WG-visible
    asm volatile("s_wait_asynccnt 0x0" ::: "memory");
    __syncthreads();

    // kick off async copy of the next chunk into the other buffer
    if (chunk + 1 < NCHUNK) {
#pragma unroll
      for (int q = 0; q < 3; ++q) {
        unsigned ldso = (unsigned)(uintptr_t)&xbuf[buf ^ 1][0] + (unsigned)(tid + q * 256) * 4u;
        unsigned voff = (unsigned)((nbase + CHUNK) * 3 + tid + q * 256) * 4u;
        asm volatile("global_load_async_to_lds_b32 %0, %1, %2 offset:0"
                     :: "v"(ldso), "v"(voff), "s"(gbase) : "memory");
      }
    }
    if (chunk + 2 < NCHUNK) {
      // warm L2 two chunks ahead (global_prefetch_b8)
      __builtin_prefetch((const void*)(xb + (size_t)(nbase + 2 * CHUNK) * 3 + tid * 3), 0, 1);
    }

    // ---- WMMA phase: 16 tiles of 16 columns; wave w does tiles w, w+8
    for (int t = wave; t < CHUNK / 16; t += 8) {
      const int colL = lane & 15;
      const int col  = t * 16 + colL;
      const int n    = nbase + col;
      const float px = xbuf[buf][col * 3 + 0];
      const float py = xbuf[buf][col * 3 + 1];
      const float pz = xbuf[buf][col * 3 + 2];
      const float nn2 = px * px + py * py + pz * pz;
      f32x2 bfrag;
      if (lane < 16) { bfrag.x = px; bfrag.y = py;   }
      else           { bfrag.x = pz; bfrag.y = 1.0f; }

      f32x8 acc = {};
      acc = __builtin_amdgcn_wmma_f32_16x16x4_f32(
          /*neg_a=*/false, afrag, /*neg_b=*/false, bfrag,
          /*c_mod=*/(short)0, acc, /*reuse_a=*/false, /*reuse_b=*/false);

      const int rbase = (lane >> 4) * 8;   // lanes 16-31 hold rows 8..15
#pragma unroll
      for (int v = 0; v < 8; ++v) {
        const int row = rbase + v;
        cd[row][KK + col] = acc[v] + nn2;
        ci[row][KK + col] = n;
      }
    }
    __syncthreads();

    // ---- selection phase: wave w merges rows 2w, 2w+1 (running-32 + 256 new)
#pragma unroll
    for (int rr = 0; rr < 2; ++rr) {
      const int row = wave * 2 + rr;
      float keep_d = FLT_BIG; int keep_i = 0;
      for (int it = 0; it < KK; ++it) {
        float best = FLT_BIG; int bslot = -1;
#pragma unroll
        for (int q = 0; q < CAND / 32; ++q) {      // 9 slots per lane
          const int slot = lane + q * 32;
          const float d = cd[row][slot];
          if (d < best) { best = d; bslot = slot; }
        }
        waveArgMin(best, bslot);
        if (it == lane) { keep_d = best; keep_i = (bslot >= 0) ? ci[row][bslot] : 0; }
        if (lane == 0 && bslot >= 0) cd[row][bslot] = FLT_BIG;   // consume
        __builtin_amdgcn_wave_barrier();
      }
      // compact: lane l holds the l-th nearest
      cd[row][lane] = keep_d;
      ci[row][lane] = keep_i;
    }
    __syncthreads();
  }

  // write knn indices: wave owns rows 2w, 2w+1
  const int r0 = wave * 2;
  knn_idx[((size_t)b * SS + s0 + r0) * KK + lane]     = ci[r0][lane];
  knn_idx[((size_t)b * SS + s0 + r0 + 1) * KK + lane] = ci[r0 + 1][lane];
}

// ---------------------------------------------------------------- 3: gather + center + stats
// one block per (b,s). Writes centered grouped [K,CH] and broadcast points [K,C]
// into final output layout; accumulates per-batch sum/sumsq for global std.
__global__ void group_kernel(const float* __restrict__ xyz,
                             const float* __restrict__ points,
                             const int* __restrict__ fps_idx,
                             const int* __restrict__ knn_idx,
                             float* __restrict__ out2,
                             float* __restrict__ stats) {
  __shared__ float g[KK][CH];
  __shared__ float meanc[CH];
  __shared__ float r1[256], r2[256];

  const size_t bs = blockIdx.x;
  const int b = (int)(bs / SS);
  const int tid = threadIdx.x;
  const int* kid = knn_idx + bs * KK;

  for (int e = tid; e < KK * CH; e += 256) {
    const int k = e / CH, c = e % CH;
    const int n = kid[k];
    float v = (c < CC) ? points[((size_t)b * NN + n) * CC + c]
                       : xyz[((size_t)b * NN + n) * 3 + (c - CC)];
    g[k][c] = v;
  }
  __syncthreads();

  if (tid < CH) {
    float mval = 0.0f;
#pragma unroll
    for (int k = 0; k < KK; ++k) mval += g[k][tid];
    meanc[tid] = mval * (1.0f / KK);
  }
  __syncthreads();

  float lsum = 0.0f, lsq = 0.0f;
  float* orow = out2 + bs * (size_t)(KK * OUTC);
  for (int e = tid; e < KK * CH; e += 256) {
    const int k = e / CH, c = e % CH;
    const float v = g[k][c] - meanc[c];
    orow[k * OUTC + c] = v;
    lsum += v; lsq += v * v;
  }

  // broadcast new_points into channels CH..CH+C-1
  const int fi = fps_idx[bs];
  for (int e = tid; e < KK * CC; e += 256) {
    const int k = e / CC, c = e % CC;
    orow[k * OUTC + CH + c] = points[((size_t)b * NN + fi) * CC + c];
  }

  r1[tid] = lsum; r2[tid] = lsq;
  __syncthreads();
#pragma unroll
  for (int off = 128; off >= 1; off >>= 1) {
    if (tid < off) { r1[tid] += r1[tid + off]; r2[tid] += r2[tid + off]; }
    __syncthreads();
  }
  if (tid == 0) {
    atomicAdd(&stats[b * 2 + 0], r1[0]);
    atomicAdd(&stats[b * 2 + 1], r2[0]);
  }
}

// ---------------------------------------------------------------- 4: normalize + affine
__global__ void norm_kernel(const float* __restrict__ alpha,
                            const float* __restrict__ beta,
                            const float* __restrict__ stats,
                            float* __restrict__ out2) {
  const size_t per_b = (size_t)SS * KK * CH;
  const size_t total = (size_t)BB * per_b;
  const float Mf = (float)per_b;
  for (size_t i = (size_t)blockIdx.x * blockDim.x + threadIdx.x; i < total;
       i += (size_t)gridDim.x * blockDim.x) {
    const size_t b = i / per_b;
    const size_t rem = i % per_b;
    const int c = (int)(rem % CH);
    const size_t sk = rem / CH;     // s*K + k
    const float sum = stats[b * 2 + 0];
    const float sq  = stats[b * 2 + 1];
    float var = (sq - sum * sum / Mf) / (Mf - 1.0f);
    float sd  = sqrtf(fmaxf(var, 0.0f)) + 1e-5f;
    float* p = out2 + (b * (size_t)SS * KK + sk) * OUTC + c;
    *p = alpha[c] * (*p / sd) + beta[c];
  }
}

// ---------------------------------------------------------------- launcher
extern "C" void kernel_launch(void* const* d_in, const int* in_sizes, int n_in,
                              void* d_out, int out_size, void* d_ws, size_t ws_size,
                              hipStream_t stream) {
  const float* xyz    = (const float*)d_in[0];
  const float* points = (const float*)d_in[1];
  const float* alpha  = (const float*)d_in[2];
  const float* beta   = (const float*)d_in[3];
  (void)in_sizes; (void)n_in; (void)out_size; (void)ws_size;

  // workspace layout
  int*   fps_idx = (int*)d_ws;                               // B*S
  int*   knn_idx = fps_idx + (size_t)BB * SS;                // B*S*K
  float* stats   = (float*)(knn_idx + (size_t)BB * SS * KK); // 2*B

  float* new_xyz = (float*)d_out;                            // B*S*3
  float* out2    = new_xyz + (size_t)BB * SS * 3;            // B*S*K*OUTC

  init_stats_kernel<<<1, 64, 0, stream>>>(stats);
  fps_kernel<<<BB, 256, 0, stream>>>(xyz, fps_idx, new_xyz);
  knn_kernel<<<dim3(SS / 16, BB), 256, 0, stream>>>(xyz, new_xyz, knn_idx);
  group_kernel<<<BB * SS, 256, 0, stream>>>(xyz, points, fps_idx, knn_idx, out2, stats);
  norm_kernel<<<8192, 256, 0, stream>>>(alpha, beta, stats, out2);
}